// MultiLoss_KLD_6579889897518
// MI455X (gfx1250) — compile-verified
//
#include <hip/hip_runtime.h>

typedef float v2f __attribute__((ext_vector_type(2)));
typedef float v8f __attribute__((ext_vector_type(8)));

#define B_ROWS   262144
#define NCOLS    99
#define ENC_COLS 12
#define NB1      1024   // pass1 blocks (1 row per thread)
#define NT1      256

// ws layout in 32-bit units
#define WS_UMIN  0      // 10 x u32 (order-encoded float min)
#define WS_UMAX  10     // 10 x u32
#define WS_CNT   20     // 2  x u32  (sex==0 count, sex==1 count)
#define WS_HIST  22     // 200 x u32 ([sex][col][bin])
#define WS_MSE   256    // 1024 x f32 per-block partials
#define WS_CE    1280   // 1024 x f32 per-block partials

// ---- order-preserving float <-> uint keys (for deterministic int atomics) ----
__device__ __forceinline__ unsigned fkey(float f) {
  unsigned u = __float_as_uint(f);
  return (u & 0x80000000u) ? ~u : (u | 0x80000000u);
}
__device__ __forceinline__ float fdec(unsigned k) {
  unsigned u = (k & 0x80000000u) ? (k ^ 0x80000000u) : ~k;
  return __uint_as_float(u);
}

// ---- numerically stable CE for one softmax block, fully unrolled ----
template<int S, int E>
__device__ __forceinline__ float ce_block(const float* __restrict__ d,
                                          const float* __restrict__ t) {
  constexpr int W = E - S;
  float v[W];
#pragma unroll
  for (int i = 0; i < W; ++i) v[i] = d[S + i];
  float m = v[0];
#pragma unroll
  for (int i = 1; i < W; ++i) m = fmaxf(m, v[i]);
  float se = 0.f, dot = 0.f;
#pragma unroll
  for (int i = 0; i < W; ++i) {
    se  += __expf(v[i] - m);
    dot += t[S + i] * v[i];
  }
  return __logf(se) + m - dot;   // logsumexp - true logit
}

// ======================= kernel 0: init ws accumulators =======================
__global__ void k_init(unsigned* __restrict__ wsU) {
  int i = threadIdx.x;
  if (i < 10) { wsU[WS_UMIN + i] = 0xFFFFFFFFu; wsU[WS_UMAX + i] = 0u; }
  if (i < 2)  { wsU[WS_CNT + i] = 0u; }
  if (i < 200) { wsU[WS_HIST + i] = 0u; }
}

// ================= kernel 1: MSE + CE partials, min/max, counts ==============
__global__ void __launch_bounds__(NT1)
k_pass1(const float* __restrict__ dec, const float* __restrict__ tru,
        const float* __restrict__ enc, unsigned* __restrict__ wsU,
        float* __restrict__ mseP, float* __restrict__ ceP) {
  __shared__ float red[NT1];
  __shared__ unsigned smin[10], smax[10], scnt[2];
  const int tid = threadIdx.x;
  if (tid < 10) { smin[tid] = 0xFFFFFFFFu; smax[tid] = 0u; }
  if (tid < 2)  { scnt[tid] = 0u; }
  __syncthreads();

  const int row = blockIdx.x * NT1 + tid;
  const float* __restrict__ d = dec + (size_t)row * NCOLS;
  const float* __restrict__ t = tru + (size_t)row * NCOLS;
  const float* __restrict__ e = enc + (size_t)row * ENC_COLS;

  // MSE over continuous cols {0,55,56,57}
  float mse = 0.f;
  {
    float a;
    a = d[0]  - t[0];  mse += a * a;
    a = d[55] - t[55]; mse += a * a;
    a = d[56] - t[56]; mse += a * a;
    a = d[57] - t[57]; mse += a * a;
  }
  // CE over the 8 one-hot blocks
  float ce = 0.f;
  ce += ce_block<1, 8 >(d, t);
  ce += ce_block<8, 24>(d, t);
  ce += ce_block<24,31>(d, t);
  ce += ce_block<31,45>(d, t);
  ce += ce_block<45,51>(d, t);
  ce += ce_block<51,53>(d, t);
  ce += ce_block<53,55>(d, t);
  ce += ce_block<58,99>(d, t);

  // encoded min/max (cols 0..9) + sex count (col 11) — integer atomics (deterministic)
#pragma unroll
  for (int i = 0; i < 10; ++i) {
    unsigned k = fkey(e[i]);
    atomicMin(&smin[i], k);
    atomicMax(&smax[i], k);
  }
  atomicAdd(&scnt[(e[11] != 0.f) ? 1 : 0], 1u);

  // fixed-tree block reductions for fp32 partials (deterministic)
  red[tid] = mse;
  __syncthreads();
  for (int s = NT1 / 2; s > 0; s >>= 1) {
    if (tid < s) red[tid] += red[tid + s];
    __syncthreads();
  }
  if (tid == 0) mseP[blockIdx.x] = red[0];
  __syncthreads();
  red[tid] = ce;
  __syncthreads();
  for (int s = NT1 / 2; s > 0; s >>= 1) {
    if (tid < s) red[tid] += red[tid + s];
    __syncthreads();
  }
  if (tid == 0) ceP[blockIdx.x] = red[0];

  if (tid < 10) {
    atomicMin(&wsU[WS_UMIN + tid], smin[tid]);
    atomicMax(&wsU[WS_UMAX + tid], smax[tid]);
  }
  if (tid < 2) atomicAdd(&wsU[WS_CNT + tid], scnt[tid]);
}

// ================= kernel 2: LDS integer histogram (L2-resident pass) ========
__global__ void __launch_bounds__(256)
k_pass2(const float* __restrict__ enc, unsigned* __restrict__ wsU) {
  __shared__ unsigned h[200];
  __shared__ float mnS[10], iwS[10];
  const int tid = threadIdx.x;
  if (tid < 200) h[tid] = 0u;
  if (tid < 10) {
    float lo = fdec(wsU[WS_UMIN + tid]);
    float hi = fdec(wsU[WS_UMAX + tid]);
    float w  = fmaxf(hi - lo, 1e-12f);
    mnS[tid] = lo;
    iwS[tid] = 10.0f / w;   // KLD_BINS / width
  }
  __syncthreads();

  const int stride = gridDim.x * blockDim.x;
  for (int row = blockIdx.x * blockDim.x + tid; row < B_ROWS; row += stride) {
    const float* __restrict__ e = enc + (size_t)row * ENC_COLS;
    unsigned* hs = &h[((e[11] != 0.f) ? 1 : 0) * 100];
#pragma unroll
    for (int i = 0; i < 10; ++i) {
      int b = (int)floorf((e[i] - mnS[i]) * iwS[i]);
      b = min(max(b, 0), 9);
      atomicAdd(&hs[i * 10 + b], 1u);
    }
  }
  __syncthreads();
  if (tid < 200) atomicAdd(&wsU[WS_HIST + tid], h[tid]);
}

// ===== WMMA-based exact f32 reduction of 1024 partials (one full wave) =======
__device__ __forceinline__ float wmma_sum1024(const float* __restrict__ p, int lane) {
  v8f acc = {};                     // C starts at 0
  v2f bones; bones.x = 1.0f; bones.y = 1.0f;   // all-ones 4x16 B
  // branch-free lane->element map for the documented 16x4 f32 A layout:
  // lanes 0-15 hold K=0,1 of row M=lane; lanes 16-31 hold K=2,3 of row M=lane-16
  const int base = (lane & 15) * 4 + (lane >> 4) * 2;
#pragma unroll
  for (int c = 0; c < 16; ++c) {    // 16 chunks x 64 values
    v2f a;
    a.x = p[c * 64 + base + 0];
    a.y = p[c * 64 + base + 1];
    // D = A x ones + C  =>  D[m][n] = sum_k A[m][k] accumulated over chunks
    acc = __builtin_amdgcn_wmma_f32_16x16x4_f32(false, a, false, bones,
                                                (short)0, acc, false, false);
  }
  // C/D layout: lane n<16 holds M=0..7 in v0..v7; lane n>=16 holds M=8..15
  float s = acc[0] + acc[1] + acc[2] + acc[3] + acc[4] + acc[5] + acc[6] + acc[7];
  return __shfl(s, 0, 32) + __shfl(s, 16, 32);
}

// ======================= kernel 3: finalize (1 wave) =========================
__global__ void __launch_bounds__(32)
k_finalize(const unsigned* __restrict__ wsU, const float* __restrict__ mseP,
           const float* __restrict__ ceP, float* __restrict__ out) {
  const int lane = threadIdx.x;
  // WMMA reductions first (EXEC all-1s, no divergence yet)
  float mseSum = wmma_sum1024(mseP, lane);
  float ceSum  = wmma_sum1024(ceP, lane);

  // KLD from integer histograms
  const float cm = fmaxf((float)wsU[WS_CNT + 0], 1.0f);
  const float cf = fmaxf((float)wsU[WS_CNT + 1], 1.0f);
  float kld = 0.f;
  for (int j = lane; j < 100; j += 32) {
    unsigned a = wsU[WS_HIST + j];         // m_hist counts
    unsigned b = wsU[WS_HIST + 100 + j];   // f_hist counts
    if (a > 0u && b > 0u) {
      float pp = (float)a / cm;
      float qq = (float)b / cf;
      kld += pp * __logf(pp / qq);
    }
  }
#pragma unroll
  for (int m = 16; m >= 1; m >>= 1) kld += __shfl_xor(kld, m, 32);

  if (lane == 0) {
    const float invB = 1.0f / (float)B_ROWS;
    float mse  = mseSum * invB;
    float ce   = ceSum * invB;
    float akld = 0.5f * kld;                   // alpha * kld
    out[0] = 0.5f * (mse + ce) + akld;         // multi_loss
    out[1] = mse;
    out[2] = ce;
    out[3] = akld;
  }
}

// =============================================================================
extern "C" void kernel_launch(void* const* d_in, const int* in_sizes, int n_in,
                              void* d_out, int out_size, void* d_ws, size_t ws_size,
                              hipStream_t stream) {
  (void)in_sizes; (void)n_in; (void)out_size; (void)ws_size;
  const float* enc = (const float*)d_in[0];   // data_encoded (B,12)
  const float* dec = (const float*)d_in[1];   // data_decoded (B,99)
  const float* tru = (const float*)d_in[2];   // data_true    (B,99)
  // d_in[3] = label_true (unused: KLD feats come from encoded cols 0..9)
  // d_in[4] = batch_size (compile-time constant here)

  unsigned* wsU = (unsigned*)d_ws;
  float*    wsF = (float*)d_ws;

  k_init<<<1, 256, 0, stream>>>(wsU);
  k_pass1<<<NB1, NT1, 0, stream>>>(dec, tru, enc, wsU, wsF + WS_MSE, wsF + WS_CE);
  k_pass2<<<512, 256, 0, stream>>>(enc, wsU);
  k_finalize<<<1, 32, 0, stream>>>(wsU, wsF + WS_MSE, wsF + WS_CE, (float*)d_out);
}